// Rotation_27315992003225
// MI455X (gfx1250) — compile-verified
//
#include <hip/hip_runtime.h>
#include <stdint.h>

#define IMG_H 512
#define IMG_W 512
#define NCH   3
#define TILE  32      // output tile edge
#define SRC_W 48      // staged source tile width  (needs 41: 32*(cos+sin)+taps+margin)
#define SRC_H 44      // staged source tile height (needs 41)
#define NTHREADS 256

typedef uint32_t u32x4 __attribute__((ext_vector_type(4)));
typedef uint32_t u32x8 __attribute__((ext_vector_type(8)));

__global__ __launch_bounds__(NTHREADS)
void rotate_bilinear_kernel(const float* __restrict__ in, float* __restrict__ out) {
    __shared__ float tile[NCH][SRC_H * SRC_W];

    const float ca = 0.96592582628906831f;   // cos(15 deg)
    const float sb = 0.25881904510252074f;   // sin(15 deg)
    const float cx = 256.0f;                 // H//2 == W//2
    const float cy = 256.0f;

    const int ox  = blockIdx.x * TILE;
    const int oy  = blockIdx.y * TILE;
    const int b   = blockIdx.z;
    const int tid = threadIdx.x;

    // ---- source bounding box of this tile (block-uniform) ----
    const float xr0 = (float)ox - cx, xr1 = (float)(ox + TILE - 1) - cx;
    const float yr0 = (float)oy - cy, yr1 = (float)(oy + TILE - 1) - cy;
    const float minxs = fminf(fminf(ca * xr0 - sb * yr0, ca * xr1 - sb * yr0),
                              fminf(ca * xr0 - sb * yr1, ca * xr1 - sb * yr1)) + cx;
    const float minys = fminf(fminf(sb * xr0 + ca * yr0, sb * xr1 + ca * yr0),
                              fminf(sb * xr0 + ca * yr1, sb * xr1 + ca * yr1)) + cy;
    const int sx0 = __builtin_amdgcn_readfirstlane((int)floorf(minxs) - 1);
    const int sy0 = __builtin_amdgcn_readfirstlane((int)floorf(minys) - 1);

    const float* base = in + (size_t)b * NCH * IMG_H * IMG_W;

    const bool interior = (sx0 >= 0) && (sy0 >= 0) &&
                          (sx0 + SRC_W <= IMG_W) && (sy0 + SRC_H <= IMG_H);

    if (interior) {
        // ================= TDM fast path: one 2D DMA per channel ==============
        if (tid < 32) {                       // one wave issues; TDM ignores EXEC
            #pragma unroll
            for (int c = 0; c < NCH; ++c) {
                const uint64_t ga = (uint64_t)(size_t)
                    (base + ((size_t)c * IMG_H + sy0) * IMG_W + sx0);
                const uint32_t lds_off = (uint32_t)(size_t)&tile[c][0];

                // ---- D# group 0 (ISA 8.3): count=1, lds_addr, global_addr, type=2
                u32x4 g0;
                g0[0] = 1u;                                        // count=1, user mode
                g0[1] = lds_off;                                   // lds_addr
                g0[2] = (uint32_t)ga;                              // global_addr[31:0]
                g0[3] = ((uint32_t)(ga >> 32) & 0x01FFFFFFu)       // global_addr[56:32]
                        | (2u << 30);                              // type=2 ("image")

                // ---- D# group 1 (ISA 8.4): data_size=4B, dims/strides in elements
                const uint32_t tdx = (uint32_t)(IMG_W - sx0);      // tensor extent from tile start
                const uint32_t tdy = (uint32_t)(IMG_H - sy0);
                u32x8 g1;
                g1[0] = (2u << 16);                 // workgroup_mask=0, data_size=2 (4B)
                g1[1] = (tdx & 0xFFFFu) << 16;      // tensor_dim0[15:0]
                g1[2] = (tdx >> 16) | ((tdy & 0xFFFFu) << 16);   // dim0 hi | dim1 lo
                g1[3] = (tdy >> 16) | ((uint32_t)SRC_W << 16);   // dim1 hi | tile_dim0
                g1[4] = (uint32_t)SRC_H;            // tile_dim1 | tile_dim2=0
                g1[5] = (uint32_t)IMG_W;            // tensor_dim0_stride[31:0] = 512
                g1[6] = 0u;                         // stride hi | tensor_dim1_stride lo
                g1[7] = 0u;

                asm volatile("tensor_load_to_lds %0, %1"
                             :: "s"(g0), "s"(g1)
                             : "memory");
            }
            __builtin_amdgcn_s_wait_tensorcnt(0);
        }
        __syncthreads();
    } else {
        // ============ border tiles: zero-fill + predicated async copy =========
        float* flat = &tile[0][0];
        for (int i = tid; i < NCH * SRC_H * SRC_W; i += NTHREADS) {
            flat[i] = 0.0f;
        }
        __syncthreads();

        for (int i = tid; i < NCH * SRC_H * SRC_W; i += NTHREADS) {
            const int c   = i / (SRC_H * SRC_W);
            const int r   = i - c * (SRC_H * SRC_W);
            const int row = r / SRC_W;
            const int col = r - row * SRC_W;
            const int gx  = sx0 + col;
            const int gy  = sy0 + row;
            if ((unsigned)gx < (unsigned)IMG_W && (unsigned)gy < (unsigned)IMG_H) {
                uint32_t lds_off = (uint32_t)(size_t)&flat[i];
                const float* gp  = base + ((size_t)c * IMG_H + gy) * IMG_W + gx;
                asm volatile("global_load_async_to_lds_b32 %0, %1, off"
                             :: "v"(lds_off), "v"(gp)
                             : "memory");
            }
        }
        asm volatile("s_wait_asynccnt 0x0" ::: "memory");
        __syncthreads();
    }

    // ---- bilinear taps from LDS, coalesced stores to global ----
    #pragma unroll
    for (int j = 0; j < (TILE * TILE) / NTHREADS; ++j) {
        const int p  = tid + NTHREADS * j;
        const int ly = p >> 5;
        const int lx = p & 31;
        const int x  = ox + lx;
        const int y  = oy + ly;

        const float xs = ca * ((float)x - cx) - sb * ((float)y - cy) + cx;
        const float ys = sb * ((float)x - cx) + ca * ((float)y - cy) + cy;
        const float fx = floorf(xs);
        const float fy = floorf(ys);
        const float wx = xs - fx;
        const float wy = ys - fy;
        const int lxs = (int)fx - sx0;   // in [1, SRC_W-2]
        const int lys = (int)fy - sy0;   // in [1, SRC_H-2]

        const float w00 = (1.0f - wy) * (1.0f - wx);
        const float w01 = (1.0f - wy) * wx;
        const float w10 = wy * (1.0f - wx);
        const float w11 = wy * wx;

        #pragma unroll
        for (int c = 0; c < NCH; ++c) {
            const float* t = &tile[c][lys * SRC_W + lxs];
            const float v = w00 * t[0] + w01 * t[1]
                          + w10 * t[SRC_W] + w11 * t[SRC_W + 1];
            out[(((size_t)b * NCH + c) * IMG_H + y) * IMG_W + x] = v;
        }
    }
}

extern "C" void kernel_launch(void* const* d_in, const int* in_sizes, int n_in,
                              void* d_out, int out_size, void* d_ws, size_t ws_size,
                              hipStream_t stream) {
    (void)n_in; (void)d_ws; (void)ws_size; (void)out_size;
    const float* in = (const float*)d_in[0];
    float* out      = (float*)d_out;
    const int batch = in_sizes[0] / (NCH * IMG_H * IMG_W);   // 32

    dim3 grid(IMG_W / TILE, IMG_H / TILE, batch);            // 16 x 16 x 32
    dim3 block(NTHREADS);
    rotate_bilinear_kernel<<<grid, block, 0, stream>>>(in, out);
}